// Point2PointLoss_38946763440607
// MI455X (gfx1250) — compile-verified
//
#include <hip/hip_runtime.h>

typedef float v2f __attribute__((ext_vector_type(2)));
typedef float v8f __attribute__((ext_vector_type(8)));

#define B_  4
#define V_  4096
#define R_  16384
#define FT_ 8192
#define RF_ 32768
#define NORMAL_THR 0.45f
#define DIST_THR   0.005f
#define KNN_WAVES  8
#define STEP_      (KNN_WAVES * 16)

__global__ void zero_f32(float* p, int n) {
  int t = blockIdx.x * blockDim.x + threadIdx.x;
  if (t < n) p[t] = 0.0f;
}

// Accumulate face normals onto vertices (faceBatchStride==0 => faces shared across batch)
__global__ void face_scatter(const float* __restrict__ verts, const int* __restrict__ faces,
                             float* __restrict__ nrm, int nF, int nV, int faceBatchStride) {
  int t = blockIdx.x * blockDim.x + threadIdx.x;
  if (t >= B_ * nF) return;
  int b = t / nF;
  int f = t - b * nF;
  const int* fp = faces + (size_t)b * faceBatchStride + (size_t)f * 3;
  int i0 = fp[0], i1 = fp[1], i2 = fp[2];
  const float* vb = verts + (size_t)b * nV * 3;
  float ax = vb[3*i0], ay = vb[3*i0+1], az = vb[3*i0+2];
  float bx = vb[3*i1], by = vb[3*i1+1], bz = vb[3*i1+2];
  float cx = vb[3*i2], cy = vb[3*i2+1], cz = vb[3*i2+2];
  float e1x = bx-ax, e1y = by-ay, e1z = bz-az;
  float e2x = cx-ax, e2y = cy-ay, e2z = cz-az;
  float nx = e1y*e2z - e1z*e2y;
  float ny = e1z*e2x - e1x*e2z;
  float nz = e1x*e2y - e1y*e2x;
  float* nb = nrm + (size_t)b * nV * 3;
  atomicAdd(&nb[3*i0+0], nx); atomicAdd(&nb[3*i0+1], ny); atomicAdd(&nb[3*i0+2], nz);
  atomicAdd(&nb[3*i1+0], nx); atomicAdd(&nb[3*i1+1], ny); atomicAdd(&nb[3*i1+2], nz);
  atomicAdd(&nb[3*i2+0], nx); atomicAdd(&nb[3*i2+1], ny); atomicAdd(&nb[3*i2+2], nz);
}

// Normalize vertex normals and emit packed points (x, y, z, |p|^2) for the KNN sweep
__global__ void normalize_and_pack(const float* __restrict__ verts, float* __restrict__ nrm,
                                   float4* __restrict__ pk, int total) {
  int t = blockIdx.x * blockDim.x + threadIdx.x;
  if (t >= total) return;
  float nx = nrm[3*t], ny = nrm[3*t+1], nz = nrm[3*t+2];
  float inv = 1.0f / fmaxf(sqrtf(nx*nx + ny*ny + nz*nz), 1e-12f);
  nrm[3*t] = nx*inv; nrm[3*t+1] = ny*inv; nrm[3*t+2] = nz*inv;
  float vx = verts[3*t], vy = verts[3*t+1], vz = verts[3*t+2];
  pk[t] = make_float4(vx, vy, vz, vx*vx + vy*vy + vz*vz);
}

// 1-NN argmin via V_WMMA_F32_16X16X4_F32 over packed float4 points.
// Block = 8 wave32s owning one 16-row query tile; waves stride the DB in 16-col chunks.
// A (16x4, K = x,y,z,0): lanes 0-15 hold K0/K1, lanes 16-31 hold K2/K3 (same row = lane&15).
// B (4x16) mirrors that layout. D[i]: lane<16 -> row i, lane>=16 -> row i+8; col = chunk + (lane&15).
// Scalar trip count keeps loop control in SALU; keys pack (clamped dist bits << 32) | col so
// v_min_u64 == argmin with first-index tie-break, matching jnp.argmin.
__global__ void __launch_bounds__(256) knn_argmin_wmma(
    const float4* __restrict__ qp, const float4* __restrict__ dbp,
    int* __restrict__ outIdx, int Nq, int Ndb) {
  int tilesPerB = Nq >> 4;
  int b    = blockIdx.x / tilesPerB;
  int tile = blockIdx.x - b * tilesPerB;
  int qbase = tile << 4;
  const float4* qb  = qp  + (size_t)b * Nq;
  const float4* dbb = dbp + (size_t)b * Ndb;

  int lane = threadIdx.x & 31;
  int wave = threadIdx.x >> 5;
  int half = lane >> 4;       // 0: K0/K1 + rows 0..7 of D; 1: K2/K3 + rows 8..15 of D
  int l16  = lane & 15;

  // A tile: one b128 load, resident in 2 VGPRs for the entire sweep
  float4 qpt = qb[qbase + l16];
  v2f a;
  a.x = half ? qpt.z : qpt.x;
  a.y = half ? 0.0f : qpt.y;

  float qnr[8];
  #pragma unroll
  for (int i = 0; i < 8; ++i) qnr[i] = qb[qbase + half * 8 + i].w;

  unsigned long long best[8];
  #pragma unroll
  for (int i = 0; i < 8; ++i) best[i] = ~0ull;

  int nIter  = Ndb / STEP_;               // exact: Ndb is 4096 or 16384
  int cstart = wave * 16;
  const float4* dptr = dbb + cstart + l16;
  float4 cur = dptr[0];
  for (int it = 0; it < nIter; ++it) {    // uniform scalar loop
    int nextOff = (it + 1 < nIter) ? (it + 1) * STEP_ : 0;  // uniform clamp on tail
    float4 nxt = dptr[nextOff];

    v2f bm;
    bm.x = half ? cur.z : cur.x;
    bm.y = half ? 0.0f : cur.y;
    float pn = cur.w;
    v8f acc = {};
    acc = __builtin_amdgcn_wmma_f32_16x16x4_f32(
        /*neg_a=*/false, a, /*neg_b=*/false, bm,
        /*c_mod=*/(short)0, acc, /*reuse_a=*/false, /*reuse_b=*/false);

    unsigned col = (unsigned)(cstart + it * STEP_ + l16);
    float d8[8];
    #pragma unroll
    for (int i = 0; i < 8; ++i)
      d8[i] = fmaxf(fmaf(-2.0f, acc[i], qnr[i] + pn), 0.0f);  // clamp: u32 order == f32 order
    #pragma unroll
    for (int i = 0; i < 8; ++i) {
      unsigned long long key = ((unsigned long long)__float_as_uint(d8[i]) << 32) | col;
      best[i] = key < best[i] ? key : best[i];
    }
    cur = nxt;
  }

  // Reduce across the 16 lanes of each half (rows i vs rows i+8 live in different halves)
  #pragma unroll
  for (int i = 0; i < 8; ++i) {
    unsigned long long k = best[i];
    #pragma unroll
    for (int m = 1; m < 16; m <<= 1) {
      unsigned long long o = __shfl_xor(k, m, 32);
      k = o < k ? o : k;
    }
    best[i] = k;
  }

  __shared__ unsigned long long red[KNN_WAVES][16];
  if (l16 == 0) {
    #pragma unroll
    for (int i = 0; i < 8; ++i) red[wave][half * 8 + i] = best[i];
  }
  __syncthreads();
  if (threadIdx.x < 16) {
    unsigned long long k = red[0][threadIdx.x];
    #pragma unroll
    for (int w = 1; w < KNN_WAVES; ++w) {
      unsigned long long o = red[w][threadIdx.x];
      k = o < k ? o : k;
    }
    outIdx[(size_t)b * Nq + qbase + threadIdx.x] = (int)(k & 0xFFFFFFFFu);
  }
}

// Masked contributions + block LDS reduction + global atomics into (B,3) output.
// Layout guarantees every 256-thread block is homogeneous in (batch, section).
__global__ void __launch_bounds__(256) finalize_loss(
    const float* __restrict__ meshes, const float* __restrict__ recons,
    const float* __restrict__ meshN, const float* __restrict__ reconN,
    const int* __restrict__ idxA, const int* __restrict__ idxB,
    const int* __restrict__ lengths, float* __restrict__ out) {
  __shared__ float sacc[3];
  if (threadIdx.x < 3) sacc[threadIdx.x] = 0.0f;
  __syncthreads();
  int t = blockIdx.x * 256 + threadIdx.x;
  int blockStart = blockIdx.x * 256;
  int b;
  float c0 = 0.0f, c1 = 0.0f, c2 = 0.0f;
  if (blockStart < B_ * V_) {           // m2r section
    b = blockStart / V_;
    int v = t - b * V_;
    const float* mb  = meshes + (size_t)b * V_ * 3;
    const float* rb  = recons + (size_t)b * R_ * 3;
    const float* mnb = meshN  + (size_t)b * V_ * 3;
    const float* rnb = reconN + (size_t)b * R_ * 3;
    int idx = idxA[(size_t)b * V_ + v];
    float dx = mb[3*v]   - rb[3*idx];
    float dy = mb[3*v+1] - rb[3*idx+1];
    float dz = mb[3*v+2] - rb[3*idx+2];
    float nx = mnb[3*v], ny = mnb[3*v+1], nz = mnb[3*v+2];
    float rnx = rnb[3*idx], rny = rnb[3*idx+1], rnz = rnb[3*idx+2];
    float nmask = (nx*rnx + ny*rny + nz*rnz >= NORMAL_THR) ? 1.0f : 0.0f;
    float dmask = (sqrtf(dx*dx + dy*dy + dz*dz) <= DIST_THR) ? 1.0f : 0.0f;
    float s = nmask * dmask * (1.0f / V_);
    c0 = fabsf(dx * rnx) * s;
    c1 = fabsf(dy * rny) * s;
    c2 = fabsf(dz * rnz) * s;
  } else {                              // r2m section
    int t2 = t - B_ * V_;
    b = (blockStart - B_ * V_) / R_;
    int r = t2 - b * R_;
    const float* mb = meshes + (size_t)b * V_ * 3;
    const float* rb = recons + (size_t)b * R_ * 3;
    int idx = idxB[(size_t)b * R_ + r];
    float dx = rb[3*r]   - mb[3*idx];
    float dy = rb[3*r+1] - mb[3*idx+1];
    float dz = rb[3*r+2] - mb[3*idx+2];
    float np    = (r < lengths[b]) ? 1.0f : 0.0f;
    float dmask = (sqrtf(dx*dx + dy*dy + dz*dz) <= DIST_THR) ? 1.0f : 0.0f;
    float s = np * dmask * (1.0f / R_);
    c0 = fabsf(dx) * s; c1 = fabsf(dy) * s; c2 = fabsf(dz) * s;
  }
  atomicAdd(&sacc[0], c0);
  atomicAdd(&sacc[1], c1);
  atomicAdd(&sacc[2], c2);
  __syncthreads();
  if (threadIdx.x < 3) atomicAdd(&out[b * 3 + threadIdx.x], sacc[threadIdx.x]);
}

extern "C" void kernel_launch(void* const* d_in, const int* in_sizes, int n_in,
                              void* d_out, int out_size, void* d_ws, size_t ws_size,
                              hipStream_t stream) {
  (void)in_sizes; (void)n_in; (void)out_size; (void)ws_size;
  // inputs: 0=joints (unused), 1=meshes, 2=recons_padded, 3=face, 4=recons_faces, 5=lengths
  const float* meshes  = (const float*)d_in[1];
  const float* recons  = (const float*)d_in[2];
  const int*   face    = (const int*)d_in[3];
  const int*   rfaces  = (const int*)d_in[4];
  const int*   lengths = (const int*)d_in[5];
  float* out = (float*)d_out;

  char* ws = (char*)d_ws;
  float*  meshN   = (float*)ws;  ws += sizeof(float)  * B_ * V_ * 3;
  float*  reconN  = (float*)ws;  ws += sizeof(float)  * B_ * R_ * 3;
  int*    idxA    = (int*)ws;    ws += sizeof(int)    * B_ * V_;
  int*    idxB    = (int*)ws;    ws += sizeof(int)    * B_ * R_;
  // float4 arrays stay 16B aligned (all preceding sizes are 16B multiples)
  float4* meshP4  = (float4*)ws; ws += sizeof(float4) * B_ * V_;
  float4* reconP4 = (float4*)ws; ws += sizeof(float4) * B_ * R_;

  int nMeshN = B_ * V_ * 3, nReconN = B_ * R_ * 3;
  zero_f32<<<(nMeshN  + 255) / 256, 256, 0, stream>>>(meshN,  nMeshN);
  zero_f32<<<(nReconN + 255) / 256, 256, 0, stream>>>(reconN, nReconN);
  zero_f32<<<1, 32, 0, stream>>>(out, B_ * 3);

  face_scatter<<<(B_ * FT_ + 255) / 256, 256, 0, stream>>>(meshes, face,   meshN,  FT_, V_, 0);
  face_scatter<<<(B_ * RF_ + 255) / 256, 256, 0, stream>>>(recons, rfaces, reconN, RF_, R_, RF_ * 3);

  normalize_and_pack<<<(B_ * V_ + 255) / 256, 256, 0, stream>>>(meshes, meshN,  meshP4,  B_ * V_);
  normalize_and_pack<<<(B_ * R_ + 255) / 256, 256, 0, stream>>>(recons, reconN, reconP4, B_ * R_);

  // Direction 1: mesh -> recon (padded recons at 9999 can never win the argmin)
  knn_argmin_wmma<<<B_ * (V_ / 16), 256, 0, stream>>>(meshP4, reconP4, idxA, V_, R_);
  // Direction 2: recon -> mesh
  knn_argmin_wmma<<<B_ * (R_ / 16), 256, 0, stream>>>(reconP4, meshP4, idxB, R_, V_);

  finalize_loss<<<(B_ * (V_ + R_)) / 256, 256, 0, stream>>>(
      meshes, recons, meshN, reconN, idxA, idxB, lengths, out);
}